// FourierDeconvolution_23450521436393
// MI455X (gfx1250) — compile-verified
//
#include <hip/hip_runtime.h>

#define TWO_PI 6.28318530717958647692f

typedef float v2f __attribute__((ext_vector_type(2)));
typedef float v8f __attribute__((ext_vector_type(8)));

__device__ __forceinline__ v8f wmma_f32(v2f a, v2f b, v8f c) {
  // V_WMMA_F32_16X16X4_F32 : D = A(16x4) * B(4x16) + C(16x16), all fp32
  return __builtin_amdgcn_wmma_f32_16x16x4_f32(
      /*neg_a=*/false, a, /*neg_b=*/false, b,
      /*c_mod=*/(short)0, c, /*reuse_a=*/false, /*reuse_b=*/false);
}

// CDNA5 async copy: 16B per lane, global -> LDS, tracked by ASYNCcnt.
// Generic pointers to __shared__ carry the LDS offset in their low 32 bits
// (flat LDS aperture: addr[31:0] = LDS offset), which is what VDST wants.
__device__ __forceinline__ void async_b128(const float* gsrc, float* ldst) {
  asm volatile("global_load_async_to_lds_b128 %0, %1, off"
               :: "v"((unsigned)(unsigned long long)ldst), "v"(gsrc)
               : "memory");
}
// Partial waits: async loads complete in-order within a wave, so waiting
// "<= n" with n = ops-in-flight-for-next-chunk guarantees the current
// chunk's copies have landed while the next chunk keeps streaming.
__device__ __forceinline__ void wait_async_le(int n) {
  if (n == 0)      asm volatile("s_wait_asynccnt 0x0" ::: "memory");
  else if (n == 6) asm volatile("s_wait_asynccnt 0x6" ::: "memory");
  else             asm volatile("s_wait_asynccnt 0x8" ::: "memory");
}

// ---------------------------------------------------------------------------
// DFT matrix: W[u][n] = exp(-2*pi*i*u*n/512) (symmetric) + negated imag plane
// so conj(F) stages are pure data copies (async-friendly).
// ---------------------------------------------------------------------------
__global__ __launch_bounds__(256)
void dft_matrix_kernel(float* __restrict__ Wr, float* __restrict__ Wi,
                       float* __restrict__ Wni) {
  int id = blockIdx.x * 256 + threadIdx.x;   // over 512*512
  int u = id >> 9;
  int n = id & 511;
  int prod = (u * n) & 511;                  // exact: twiddle period is 512
  float s, c;
  __sincosf(-TWO_PI * (float)prod * (1.0f / 512.0f), &s, &c);
  Wr[id] = c;
  Wi[id] = s;
  Wni[id] = -s;
}

// ---------------------------------------------------------------------------
// Dk = psf2otf(k, (512,512)), k is (B,1,25,25), circular shift by (12,12).
// Incremental complex-rotation twiddles (2 sincos per axis per thread).
// ---------------------------------------------------------------------------
__global__ __launch_bounds__(256)
void dk_otf_kernel(const float* __restrict__ kpsf,
                   float* __restrict__ DKr, float* __restrict__ DKi) {
  int id  = blockIdx.x * 256 + threadIdx.x;  // over 4*512*512
  int b   = id >> 18;
  int rem = id & 262143;
  int u   = rem >> 9;
  int v   = rem & 511;
  const float* kb = kpsf + b * 625;

  float stepys, stepyc, stepxs, stepxc;
  __sincosf(-TWO_PI * (float)u * (1.0f / 512.0f), &stepys, &stepyc);
  __sincosf(-TWO_PI * (float)v * (1.0f / 512.0f), &stepxs, &stepxc);
  float pyr, pyi, px0r, px0i;
  __sincosf(TWO_PI * (float)u * (12.0f / 512.0f), &pyi, &pyr);   // ky = 0 term
  __sincosf(TWO_PI * (float)v * (12.0f / 512.0f), &px0i, &px0r); // kx = 0 term

  float accR = 0.0f, accI = 0.0f;
  for (int ky = 0; ky < 25; ky++) {
    float pxr = px0r, pxi = px0i;
    #pragma unroll 5
    for (int kx = 0; kx < 25; kx++) {
      float kv = kb[ky * 25 + kx];
      accR += kv * (pyr * pxr - pyi * pxi);
      accI += kv * (pyr * pxi + pyi * pxr);
      float nr = pxr * stepxc - pxi * stepxs;
      float ni = pxr * stepxs + pxi * stepxc;
      pxr = nr; pxi = ni;
    }
    float nr = pyr * stepyc - pyi * stepys;
    float ni = pyr * stepys + pyi * stepyc;
    pyr = nr; pyi = ni;
  }
  DKr[id] = accR;
  DKi[id] = accI;
}

// ---------------------------------------------------------------------------
// DG[c,u,v] = sum_f |psf2otf(weights[f,c], (512,512))|^2 , shift (2,2)
// ---------------------------------------------------------------------------
__global__ __launch_bounds__(256)
void dg_sum_kernel(const float* __restrict__ wts, float* __restrict__ DG) {
  int id  = blockIdx.x * 256 + threadIdx.x;  // over 3*512*512
  int c   = id >> 18;
  int rem = id & 262143;
  int u   = rem >> 9;
  int v   = rem & 511;

  float cyr[5], cyi[5], cxr[5], cxi[5];
  #pragma unroll
  for (int j = 0; j < 5; j++) {
    __sincosf(-TWO_PI * (float)u * (float)(j - 2) * (1.0f / 512.0f), &cyi[j], &cyr[j]);
    __sincosf(-TWO_PI * (float)v * (float)(j - 2) * (1.0f / 512.0f), &cxi[j], &cxr[j]);
  }
  float dg = 0.0f;
  for (int f = 0; f < 24; f++) {
    const float* wf = wts + (f * 3 + c) * 25;
    float aR = 0.0f, aI = 0.0f;
    #pragma unroll
    for (int j = 0; j < 5; j++) {
      #pragma unroll
      for (int i = 0; i < 5; i++) {
        float wv = wf[j * 5 + i];
        aR += wv * (cyr[j] * cxr[i] - cyi[j] * cxi[i]);
        aI += wv * (cyr[j] * cxi[i] + cyi[j] * cxr[i]);
      }
    }
    dg += aR * aR + aI * aI;
  }
  DG[id] = dg;
}

// ---------------------------------------------------------------------------
// Batched 512^3 complex GEMM via Karatsuba (3 real WMMA per complex k-step):
//   P1 = Ar*Br, P2 = Ai*Bi, P3 = (Ar+Ai)*(Br+Bi)
//   Cre = P1 - P2, Cim = P3 - P1 - P2
// REAL_A: A imag plane is identically zero -> skip P2 and lAi entirely.
// MODE 0: write complex C
// MODE 1: write real(C)*scale only      (final ifft2 + real())
// MODE 2: fused Wiener epilogue: acc = Fy, C holds fft2(phi) on input;
//         writes num = Omega * (conj(Dk)*Fy + fft2(phi)/w) back into C.
// Tiles streamed with GLOBAL_LOAD_ASYNC_TO_LDS_B128 (ASYNCcnt), double-
// buffered in LDS so chunk k+1's DMA overlaps chunk k's WMMA.
// Block: 64x64 tile, 256 threads = 8 wave32; each wave -> two 16x16 tiles.
// ---------------------------------------------------------------------------
template <int MODE, bool REAL_A>
__global__ __launch_bounds__(256)
void zgemm512(const float* __restrict__ Ar, const float* __restrict__ Ai, int sA,
              const float* __restrict__ Br, const float* __restrict__ Bi, int sB,
              float* __restrict__ Cr, float* __restrict__ Ci, int sC,
              float scale,
              const float* __restrict__ DKr, const float* __restrict__ DKi,
              const float* __restrict__ DG, const float* __restrict__ wvec) {
  const int N = 512;
  const int HW = 512 * 512;
  const int PER_CHUNK = REAL_A ? 6 : 8;  // async instrs in flight per chunk
  // Row pitches are multiples of 16B (B128 LDS alignment) and keep the
  // A-fragment reads conflict-free (gcd(36,64)=4 -> 16 distinct banks).
  __shared__ __align__(16) float lAr[2][64][36];
  __shared__ __align__(16) float lAi[REAL_A ? 1 : 2][REAL_A ? 1 : 64][36];
  __shared__ __align__(16) float lBr[2][32][68];
  __shared__ __align__(16) float lBi[2][32][68];

  int tid  = threadIdx.x;
  int lane = tid & 31;
  int wv   = tid >> 5;
  int wm   = (wv & 1) * 32;   // wave M offset inside block tile
  int wn   = (wv >> 1) * 16;  // wave N offset inside block tile
  int z    = blockIdx.z;
  int rowBase = blockIdx.y * 64;
  int colBase = blockIdx.x * 64;

  const float* arp = Ar + (size_t)z * (size_t)sA;
  const float* aip = REAL_A ? nullptr : (Ai + (size_t)z * (size_t)sA);
  const float* brp = Br + (size_t)z * (size_t)sB;
  const float* bip = Bi + (size_t)z * (size_t)sB;

  // Issue one chunk's async copies into LDS buffer pb.
  auto stage = [&](int k0, int pb) {
    #pragma unroll
    for (int t = 0; t < 2; t++) {
      int q  = tid + t * 256;          // 512 x 16B chunks per plane
      int rA = q >> 3, cA = (q & 7) << 2;    // A: 64 rows x 8 chunks
      async_b128(arp + (size_t)(rowBase + rA) * N + (k0 + cA), &lAr[pb][rA][cA]);
      if constexpr (!REAL_A)
        async_b128(aip + (size_t)(rowBase + rA) * N + (k0 + cA), &lAi[pb][rA][cA]);
      int rB = q >> 4, cB = (q & 15) << 2;   // B: 32 rows x 16 chunks
      async_b128(brp + (size_t)(k0 + rB) * N + (colBase + cB), &lBr[pb][rB][cB]);
      async_b128(bip + (size_t)(k0 + rB) * N + (colBase + cB), &lBi[pb][rB][cB]);
    }
  };

  v8f acc1[2] = {}, acc2[2] = {}, acc3[2] = {};

  stage(0, 0);  // prologue: start chunk 0 DMA
  int p = 0;
  for (int k0 = 0; k0 < N; k0 += 32, p ^= 1) {
    __syncthreads();  // all waves done reading buffer p^1 (chunk k-1)
    if (k0 + 32 < N) stage(k0 + 32, p ^ 1);           // overlap next DMA
    wait_async_le((k0 + 32 < N) ? PER_CHUNK : 0);     // chunk k landed
    __syncthreads();  // chunk k visible to all waves

    // ---- compute: 8 complex k-steps of K=4 on buffer p ----
    int bl = lane & 15;
    int bh = lane >> 4;
    #pragma unroll
    for (int kk = 0; kk < 32; kk += 4) {
      // B fragment (4x16): v0 = rows k0,k1 ; v1 = rows k2,k3 (half-wave split)
      v2f brf, bif;
      brf.x = lBr[p][kk + bh][wn + bl];      brf.y = lBr[p][kk + 2 + bh][wn + bl];
      bif.x = lBi[p][kk + bh][wn + bl];      bif.y = lBi[p][kk + 2 + bh][wn + bl];
      v2f bsf = brf + bif;
      int kq = kk + (bh << 1);
      #pragma unroll
      for (int s = 0; s < 2; s++) {
        int arow = wm + s * 16 + bl;
        v2f arf;
        arf.x = lAr[p][arow][kq];  arf.y = lAr[p][arow][kq + 1];
        if constexpr (REAL_A) {
          acc1[s] = wmma_f32(arf, brf, acc1[s]);   // P1
          acc3[s] = wmma_f32(arf, bsf, acc3[s]);   // P3 (Ai == 0)
        } else {
          v2f aif;
          aif.x = lAi[p][arow][kq];  aif.y = lAi[p][arow][kq + 1];
          v2f asf = arf + aif;
          acc1[s] = wmma_f32(arf, brf, acc1[s]);   // P1
          acc2[s] = wmma_f32(aif, bif, acc2[s]);   // P2
          acc3[s] = wmma_f32(asf, bsf, acc3[s]);   // P3
        }
      }
    }
  }

  // Epilogue. C/D layout: lanes 0-15 -> M=0..7 (v0..v7), lanes 16-31 -> M=8..15
  int col = colBase + wn + (lane & 15);
  int rb0 = rowBase + wm + (lane >> 4) * 8;
  int b = z / 3, cch = z % 3;

  #pragma unroll
  for (int s = 0; s < 2; s++) {
    #pragma unroll
    for (int r = 0; r < 8; r++) {
      float p1 = acc1[s][r];
      float p3 = acc3[s][r];
      float re, im;
      if constexpr (REAL_A) {
        re = p1;
        im = p3 - p1;
      } else {
        float p2 = acc2[s][r];
        re = p1 - p2;
        im = p3 - p1 - p2;
      }
      int row = rb0 + s * 16 + r;
      size_t cidx = (size_t)z * (size_t)sC + (size_t)row * N + col;
      if constexpr (MODE == 0) {
        Cr[cidx] = re;
        Ci[cidx] = im;
      } else if constexpr (MODE == 1) {
        Cr[cidx] = re * scale;
      } else { // MODE 2: fused Wiener combine (1/w folded in here by linearity)
        size_t pp = (size_t)row * N + col;
        float dr = DKr[(size_t)b * HW + pp];
        float di = DKi[(size_t)b * HW + pp];
        float dg = DG[(size_t)cch * HW + pp];
        float invw = 1.0f / wvec[b];
        float omega = 1.0f / (dr * dr + di * di + dg * invw);
        float nr = dr * re + di * im + invw * Cr[cidx]; // conj(Dk)*Fy + Fphi/w
        float ni = dr * im - di * re + invw * Ci[cidx];
        Cr[cidx] = omega * nr;
        Ci[cidx] = omega * ni;
      }
    }
  }
}

// ---------------------------------------------------------------------------
// Host-side orchestration (all on `stream`, graph-capture safe)
// ---------------------------------------------------------------------------
extern "C" void kernel_launch(void* const* d_in, const int* in_sizes, int n_in,
                              void* d_out, int out_size, void* d_ws, size_t ws_size,
                              hipStream_t stream) {
  const float* y   = (const float*)d_in[0];  // (4,3,512,512)
  const float* k   = (const float*)d_in[1];  // (4,1,25,25)
  const float* w   = (const float*)d_in[2];  // (4,1)
  const float* phi = (const float*)d_in[3];  // (4,3,512,512)
  const float* wts = (const float*)d_in[4];  // (24,3,5,5)
  (void)in_sizes; (void)n_in; (void)out_size; (void)ws_size;

  const size_t HW = 512 * 512;
  float* ws  = (float*)d_ws;
  float* Wr  = ws;                 // DFT matrix real
  float* Wi  = Wr + HW;            // DFT matrix imag
  float* Wni = Wi + HW;            // DFT matrix imag, negated (conj(F))
  float* DKr = Wni + HW;           // Dk real   [4,HW]
  float* DKi = DKr + 4 * HW;       // Dk imag   [4,HW]
  float* DG  = DKi + 4 * HW;       // sum|Dg|^2 [3,HW]
  float* bAr = DG + 3 * HW;        // scratch complex A [12,HW]
  float* bAi = bAr + 12 * HW;
  float* bBr = bAi + 12 * HW;      // scratch complex B [12,HW]
  float* bBi = bBr + 12 * HW;

  dft_matrix_kernel<<<(int)(HW / 256), 256, 0, stream>>>(Wr, Wi, Wni);
  dk_otf_kernel<<<(int)(4 * HW / 256), 256, 0, stream>>>(k, DKr, DKi);
  dg_sum_kernel<<<(int)(3 * HW / 256), 256, 0, stream>>>(wts, DG);

  dim3 g(8, 8, 12), blk(256);
  const float invN2 = 1.0f / (512.0f * 512.0f);

  // 1) T = phi * F                 -> bA   (1/w deferred to stage 4 epilogue)
  zgemm512<0, true><<<g, blk, 0, stream>>>(phi, nullptr, (int)HW, Wr, Wi, 0,
                                           bAr, bAi, (int)HW, 1.f,
                                           nullptr, nullptr, nullptr, nullptr);
  // 2) fft2(phi) = F * T           -> bB
  zgemm512<0, false><<<g, blk, 0, stream>>>(Wr, Wi, 0, bAr, bAi, (int)HW,
                                            bBr, bBi, (int)HW, 1.f,
                                            nullptr, nullptr, nullptr, nullptr);
  // 3) T = y * F                   -> bA
  zgemm512<0, true><<<g, blk, 0, stream>>>(y, nullptr, (int)HW, Wr, Wi, 0,
                                           bAr, bAi, (int)HW, 1.f,
                                           nullptr, nullptr, nullptr, nullptr);
  // 4) Fy = F * T, fused: num = Omega*(conj(Dk)*Fy + fft2(phi)/w) -> bB
  zgemm512<2, false><<<g, blk, 0, stream>>>(Wr, Wi, 0, bAr, bAi, (int)HW,
                                            bBr, bBi, (int)HW, 1.f,
                                            DKr, DKi, DG, w);
  // 5) T2 = num * conj(F)          -> bA
  zgemm512<0, false><<<g, blk, 0, stream>>>(bBr, bBi, (int)HW, Wr, Wni, 0,
                                            bAr, bAi, (int)HW, 1.f,
                                            nullptr, nullptr, nullptr, nullptr);
  // 6) out = Re(conj(F) * T2) / N^2  -> d_out
  zgemm512<1, false><<<g, blk, 0, stream>>>(Wr, Wni, 0, bAr, bAi, (int)HW,
                                            (float*)d_out, nullptr, (int)HW, invN2,
                                            nullptr, nullptr, nullptr, nullptr);
}